// LinearFunctionals2d_41102837023270
// MI455X (gfx1250) — compile-verified
//
#include <hip/hip_runtime.h>
#include <hip/hip_bf16.h>

// LinearFunctionals2d: truncated-spectrum FNO functional as 3 f32-WMMA GEMMs.
// B=32, CI=32, CO=64, NX=NY=256, M1=M2=32.
//
//  Stage1: S1[(b,i,nx), 2*ky+c]   = X(262144x256) * Fy(256x66)      (DFT over ny, norm 1/65536)
//  Stage2: A3[b, (i,kxIdx,ky,c)]  = s * [ S1-rows(33792x512) * B2(512x128) ]  (DFT over nx, kept rows)
//  Stage3: out(32x64)            += A3(32x135168) * [Wr | -Wi](135168x64)     (K-split + atomics)
//
// All matrix math uses V_WMMA_F32_16X16X4_F32 (wave32, A/B=2 VGPR, C/D=8 VGPR).
// Twiddle tables are stored ROW-PAIR INTERLEAVED so each B fragment is one
// global_load_b64 (rows k and k+1 adjacent), matching the b64 A-fragment loads.

typedef __attribute__((ext_vector_type(2))) float v2f;
typedef __attribute__((ext_vector_type(8))) float v8f;

#define NXC 256
#define NYC 256
#define FY_LD 80          // padded N for stage1 (66 valid)
#define S1_COLS 66
#define R1 262144         // B*CI*NX
#define R2 33792          // B*CI*(M2+1)
#define K3 135168         // CI*64*33*2
#define FY_ELEMS (256*FY_LD)            // 20480 floats
#define B2_ELEMS (512*128)              // 65536 floats

static __device__ __forceinline__ v8f wmma4(v2f a, v2f b, v8f c) {
    return __builtin_amdgcn_wmma_f32_16x16x4_f32(false, a, false, b, (short)0, c, false, false);
}

// ------ twiddle init: Fy2 (128 pairs x 80) and B2p (256 pairs x 128) ------
// Pair-interleaved: T[(k2*LD + col)*2 + c] holds original row k = 2*k2 + c.
__global__ void fno_init_tw(float* __restrict__ Fy2, float* __restrict__ B2p) {
    int idx = blockIdx.x * 256 + threadIdx.x;           // exactly 86016 threads
    const float STEP = 6.28318530717958647692f / 256.0f;
    if (idx < FY_ELEMS) {
        int c  = idx & 1;
        int col = (idx >> 1) % FY_LD;
        int k2  = (idx >> 1) / FY_LD;
        int ny  = 2 * k2 + c;                           // original K row (ny)
        float v = 0.0f;
        if (col < S1_COLS) {
            int ky = col >> 1, co = col & 1;
            int m = (ky * ny) & 255;                    // exact phase reduction
            float th = STEP * (float)m;
            v = (co ? -__sinf(th) : __cosf(th)) * (1.0f / 65536.0f);
        }
        Fy2[idx] = v;
    } else {
        int j  = idx - FY_ELEMS;
        int ci = j & 1;                                 // input re/im (K row parity)
        int col = (j >> 1) & 127;
        int nx  = (j >> 1) >> 7;                        // pair index == nx
        int kxIdx = col >> 1, co = col & 1;             // output re/im
        int kx = (kxIdx < 32) ? kxIdx : (192 + kxIdx);  // 224..255 for conj block
        int m = (kx * nx) & 255;
        float th = STEP * (float)m;
        float tr = __cosf(th), ti = -__sinf(th);        // e^{-i th}
        // (ar + i*ai)*(tr + i*ti): re = ar*tr - ai*ti ; im = ar*ti + ai*tr
        B2p[j] = (ci == 0) ? (co == 0 ? tr : ti) : (co == 0 ? -ti : tr);
    }
}

// ---------------- stage 1: DFT over ny ------------------------------------
// grid: 2048 blocks x 256 thr (8 waves); each wave: 16-row tile, full width.
__global__ void fno_s1_dfty(const float* __restrict__ x, const float* __restrict__ Fy2,
                            float* __restrict__ S1) {
    const int lane = threadIdx.x & 31;
    const int wave = threadIdx.x >> 5;
    const int r0 = (blockIdx.x * 8 + wave) * 16;
    const int m  = lane & 15;
    const int k0 = (lane < 16) ? 0 : 2;                 // A/B fragment K split (even)

    v8f acc[5];
    #pragma unroll
    for (int t = 0; t < 5; ++t) acc[t] = (v8f){0,0,0,0,0,0,0,0};

    const float* arow = x + (size_t)(r0 + m) * NYC + k0;
    const float* bbase = Fy2 + ((size_t)(k0 >> 1) * FY_LD + m) * 2;
    for (int kb = 0; kb < NYC; kb += 4) {
        v2f a = *reinterpret_cast<const v2f*>(arow + kb);          // A[m, k..k+1]
        const float* bp = bbase + (size_t)(kb >> 1) * (2 * FY_LD);
        #pragma unroll
        for (int t = 0; t < 5; ++t) {
            v2f b = *reinterpret_cast<const v2f*>(bp + t * 32);    // B[k][n],B[k+1][n]
            acc[t] = wmma4(a, b, acc[t]);
        }
    }
    const int rowAdd = (lane < 16) ? 0 : 8;                        // C/D layout
    #pragma unroll
    for (int t = 0; t < 5; ++t) {
        int col = t * 16 + m;
        if (col < S1_COLS) {
            #pragma unroll
            for (int v = 0; v < 8; ++v)
                S1[(size_t)(r0 + v + rowAdd) * S1_COLS + col] = acc[t][v];
        }
    }
}

// ---------------- stage 2: DFT over nx on kept rows, apply mode weight s --
// rows r = (b*32+i)*33 + ky ; K = 2*nx+c ; N = 2*kxIdx+c. grid: 264 x 256.
__global__ void fno_s2_dftx(const float* __restrict__ S1, const float* __restrict__ B2p,
                            float* __restrict__ A3) {
    const int lane = threadIdx.x & 31;
    const int wave = threadIdx.x >> 5;
    const int r0 = (blockIdx.x * 8 + wave) * 16;
    const int m  = lane & 15;
    const int k0 = (lane < 16) ? 0 : 2;
    const int row = r0 + m;
    const int bi = row / 33;                 // b*32+i
    const int ky = row % 33;
    const float* abase = S1 + (size_t)bi * (NXC * S1_COLS) + ky * 2;
    const float* bbase = B2p + ((size_t)(k0 >> 1) * 128 + m) * 2;

    v8f acc[8];
    #pragma unroll
    for (int t = 0; t < 8; ++t) acc[t] = (v8f){0,0,0,0,0,0,0,0};

    for (int kb = 0; kb < 512; kb += 4) {
        int nx = (kb + k0) >> 1;                                   // k even
        v2f a = *reinterpret_cast<const v2f*>(abase + nx * S1_COLS); // (re,im)
        const float* bp = bbase + (size_t)(kb >> 1) * 256;
        #pragma unroll
        for (int t = 0; t < 8; ++t) {
            v2f b = *reinterpret_cast<const v2f*>(bp + t * 32);
            acc[t] = wmma4(a, b, acc[t]);
        }
    }
    const int rowAdd = (lane < 16) ? 0 : 8;
    #pragma unroll
    for (int t = 0; t < 8; ++t) {
        int col = t * 16 + m;
        int kxIdx = col >> 1, c = col & 1;
        #pragma unroll
        for (int v = 0; v < 8; ++v) {
            int rr  = r0 + v + rowAdd;
            int kyy = rr % 33;
            float s = 2.0f;
            if (kxIdx >= 32 && kyy == 0) s = 0.0f;   // conj block, ky=0 excluded
            if (kxIdx == 0  && kyy == 0) s = 1.0f;   // DC: 2 - 1
            A3[(size_t)((rr / 33) * 64 + kxIdx) * 66 + kyy * 2 + c] = s * acc[t][v];
        }
    }
}

// ---------------- stage 3: channel mix + mode sum, K-split ----------------
// out(32x64) = A3(32x135168) * [Wr | -Wi]. 8 tiles * 256 K-chunks = 2048 waves.
__global__ void fno_s3_mix(const float* __restrict__ A3, const float* __restrict__ wr,
                           const float* __restrict__ wi, float* __restrict__ out) {
    const int lane  = threadIdx.x & 31;
    const int task  = blockIdx.x * 8 + (threadIdx.x >> 5);   // 0..2047
    const int tile  = task >> 8;                             // 0..7
    const int chunk = task & 255;
    const int rt = tile >> 2, nt = tile & 3;
    const int m  = lane & 15;
    const int k0 = (lane < 16) ? 0 : 2;
    const int o  = nt * 16 + m;
    const float* abase = A3 + (size_t)(rt * 16 + m) * K3;

    v8f acc = (v8f){0,0,0,0,0,0,0,0};
    const int kstart = chunk * 528;           // 256 chunks * 528 = 135168
    for (int kb = kstart; kb < kstart + 528; kb += 4) {
        int k = kb + k0;                      // even: k -> (re, Wr), k+1 -> (im, -Wi)
        v2f a = *reinterpret_cast<const v2f*>(abase + k);
        int i    = k / 4224;                  // 64*33*2
        int rem  = k - i * 4224;
        int kxI  = rem / 66;
        int rem2 = rem - kxI * 66;
        int kyy  = rem2 >> 1;                 // rem2 even here
        size_t widx = ((size_t)(i * 64 + o) * 64 + kxI) * 33 + kyy;
        v2f b; b.x = wr[widx]; b.y = -wi[widx];
        acc = wmma4(a, b, acc);
    }
    const int rowAdd = (lane < 16) ? 0 : 8;
    #pragma unroll
    for (int v = 0; v < 8; ++v)
        atomicAdd(&out[(rt * 16 + v + rowAdd) * 64 + o], acc[v]);
}

extern "C" void kernel_launch(void* const* d_in, const int* in_sizes, int n_in,
                              void* d_out, int out_size, void* d_ws, size_t ws_size,
                              hipStream_t stream) {
    (void)in_sizes; (void)n_in; (void)ws_size;
    const float* x  = (const float*)d_in[0];
    const float* wr = (const float*)d_in[1];
    const float* wi = (const float*)d_in[2];
    float* out = (float*)d_out;

    float* ws  = (float*)d_ws;
    float* Fy2 = ws;                                  // 20480 f   (pair-interleaved)
    float* B2p = ws + FY_ELEMS;                       // 65536 f   (pair-interleaved)
    float* S1  = ws + FY_ELEMS + B2_ELEMS;            // 262144*66 f (~69 MB)
    float* A3  = S1 + (size_t)R1 * S1_COLS;           // 32*135168 f (~17 MB)

    hipMemsetAsync(d_out, 0, (size_t)out_size * sizeof(float), stream);
    fno_init_tw<<<(FY_ELEMS + B2_ELEMS) / 256, 256, 0, stream>>>(Fy2, B2p);
    fno_s1_dfty<<<R1 / (16 * 8), 256, 0, stream>>>(x, Fy2, S1);
    fno_s2_dftx<<<R2 / (16 * 8), 256, 0, stream>>>(S1, B2p, A3);
    fno_s3_mix<<<2048 / 8, 256, 0, stream>>>(A3, wr, wi, out);
}